// Model1_40759239639954
// MI455X (gfx1250) — compile-verified
//
#include <hip/hip_runtime.h>
#include <math.h>

#define NNODES 200000
#define NEDGES 3200000
#define NGRAPH 128
#define FIN    128
#define NCLS   10
#define KSEL   30
#define DCAT   97

typedef __attribute__((ext_vector_type(2))) float v2f;
typedef __attribute__((ext_vector_type(8))) float v8f;

// ---------------------------------------------------------------- utilities
__global__ void fill_f32(float* __restrict__ p, float v, int n) {
    int i = blockIdx.x * blockDim.x + threadIdx.x;
    if (i < n) p[i] = v;
}

// deg starts at 1.0 (self-loop term); add 1 per non-self incoming edge
__global__ void deg_accum(const int* __restrict__ src, const int* __restrict__ dst,
                          float* __restrict__ deg) {
    int e = blockIdx.x * blockDim.x + threadIdx.x;
    if (e >= NEDGES) return;
    int s = src[e], d = dst[e];
    if (s != d) atomicAdd(&deg[d], 1.0f);
}

// dinv = rsqrt(deg); deg buffer becomes self_norm = 1/deg (in place)
__global__ void norms_k(float* __restrict__ deg, float* __restrict__ dinv) {
    int i = blockIdx.x * blockDim.x + threadIdx.x;
    if (i >= NNODES) return;
    float dg = deg[i];
    dinv[i] = rsqrtf(dg);
    deg[i]  = 1.0f / dg;
}

__global__ void enorm_k(const int* __restrict__ src, const int* __restrict__ dst,
                        const float* __restrict__ dinv, float* __restrict__ enorm) {
    int e = blockIdx.x * blockDim.x + threadIdx.x;
    if (e >= NEDGES) return;
    int s = src[e], d = dst[e];
    enorm[e] = (s != d) ? dinv[s] * dinv[d] : 0.0f;
}

// ------------------------------------------------- fp32 WMMA GEMM  (X @ W)
// X: NNODES x Kdim (row stride in_stride), W: Kdim x 32 row-major.
// Writes h = X@W (N x 32, dense) and out = h*self_norm + bias (stride out_stride).
//
// 64-thread block = 2 waves; each wave owns 32 rows (two 16-row tiles) and the
// full 32-wide output (two 16-col tiles) -> 4 v8f accumulators, 4 WMMAs per
// k-step. W is staged in LDS pair-interleaved so each lane's B fragment
// (W[kk][n], W[kk+1][n]) is ONE aligned ds b64 load (no repacking movs).
__global__ __launch_bounds__(64)
void gemm_wmma_f32(const float* __restrict__ X, int in_stride,
                   const float* __restrict__ W, int Kdim,
                   const float* __restrict__ bias,
                   const float* __restrict__ self_norm,
                   float* __restrict__ h,
                   float* __restrict__ out, int out_stride) {
    __shared__ float sW[FIN * 32];                 // packed: [(k/2)*64 + 2n + (k&1)]
    const int tid = threadIdx.x;
    for (int i = tid; i < Kdim * 32; i += 64) {
        const int k = i >> 5, n = i & 31;
        sW[((k >> 1) * 32 + n) * 2 + (k & 1)] = W[i];
    }
    __syncthreads();

    const int wave = tid >> 5;
    const int lane = tid & 31;
    const int lo   = lane & 15;                    // column / row-within-half
    const int hi   = lane >> 4;                    // half-wave: K sub-pair select
    const int m0   = (blockIdx.x * 2 + wave) * 32; // 200000 = 3125*64 exact

    v8f acc00 = {}, acc01 = {};                    // rows m0..m0+15, cols 0-15/16-31
    v8f acc10 = {}, acc11 = {};                    // rows m0+16..m0+31
    const float* xr0 = X + (size_t)(m0 + lo) * in_stride;
    const float* xr1 = X + (size_t)(m0 + 16 + lo) * in_stride;

    for (int k = 0; k < Kdim; k += 4) {
        const int kk = k + 2 * hi;
        // A 16x4 fp32: lanes 0-15 hold K=k,k+1; lanes 16-31 hold K=k+2,k+3
        v2f a0, a1;
        a0.x = xr0[kk]; a0.y = xr0[kk + 1];
        a1.x = xr1[kk]; a1.y = xr1[kk + 1];
        // B 4x16 fp32, pair-packed: one b64 LDS load per fragment
        const int pb = (kk >> 1) * 64;
        const v2f b0 = *(const v2f*)&sW[pb + 2 * lo];
        const v2f b1 = *(const v2f*)&sW[pb + 32 + 2 * lo];
        acc00 = __builtin_amdgcn_wmma_f32_16x16x4_f32(false, a0, false, b0,
                                                      (short)0, acc00, false, false);
        acc01 = __builtin_amdgcn_wmma_f32_16x16x4_f32(false, a0, false, b1,
                                                      (short)0, acc01, false, false);
        acc10 = __builtin_amdgcn_wmma_f32_16x16x4_f32(false, a1, false, b0,
                                                      (short)0, acc10, false, false);
        acc11 = __builtin_amdgcn_wmma_f32_16x16x4_f32(false, a1, false, b1,
                                                      (short)0, acc11, false, false);
    }

    // D layout: VGPR v -> row base+v (lanes 0-15) / base+8+v (lanes 16-31), col lo
    const float bl = bias[lo];
    const float bh = bias[16 + lo];
#pragma unroll
    for (int v = 0; v < 8; ++v) {
        const int mA = m0 + v + 8 * hi;
        const int mB = mA + 16;
        const float snA = self_norm[mA];
        const float snB = self_norm[mB];
        h[(size_t)mA * 32 + lo]      = acc00[v];
        h[(size_t)mA * 32 + 16 + lo] = acc01[v];
        h[(size_t)mB * 32 + lo]      = acc10[v];
        h[(size_t)mB * 32 + 16 + lo] = acc11[v];
        out[(size_t)mA * out_stride + lo]      = acc00[v] * snA + bl;
        out[(size_t)mA * out_stride + 16 + lo] = acc01[v] * snA + bh;
        out[(size_t)mB * out_stride + lo]      = acc10[v] * snB + bl;
        out[(size_t)mB * out_stride + 16 + lo] = acc11[v] * snB + bh;
    }
}

// ------------------------------------------------ edge scatter (32 features)
// thread = (edge, feature); 32 lanes cover one contiguous h row -> coalesced
// loads and 32 consecutive global_atomic_add_f32 that resolve in L2.
__global__ void scatter32(const int* __restrict__ src, const int* __restrict__ dst,
                          const float* __restrict__ enorm,
                          const float* __restrict__ h,
                          float* __restrict__ out, int out_stride) {
    long long t = (long long)blockIdx.x * blockDim.x + threadIdx.x;
    if (t >= (long long)NEDGES * 32) return;
    int e = (int)(t >> 5);
    int f = (int)(t & 31);
    float en = enorm[e];
    if (en == 0.0f) return;
    int s = src[e], d = dst[e];
    atomicAdd(&out[(size_t)d * out_stride + f], en * h[(size_t)s * 32 + f]);
}

__global__ void scatter1(const int* __restrict__ src, const int* __restrict__ dst,
                         const float* __restrict__ enorm,
                         const float* __restrict__ h4,
                         float* __restrict__ out, int out_stride) {
    int e = blockIdx.x * blockDim.x + threadIdx.x;
    if (e >= NEDGES) return;
    float en = enorm[e];
    if (en == 0.0f) return;
    atomicAdd(&out[(size_t)dst[e] * out_stride], en * h4[src[e]]);
}

__global__ void tanh32(float* __restrict__ p, int stride) {
    long long t = (long long)blockIdx.x * blockDim.x + threadIdx.x;
    if (t >= (long long)NNODES * 32) return;
    int i = (int)(t >> 5), f = (int)(t & 31);
    float* q = &p[(size_t)i * stride + f];
    *q = tanhf(*q);
}

__global__ void tanh1(float* __restrict__ p, int stride) {
    int i = blockIdx.x * blockDim.x + threadIdx.x;
    if (i >= NNODES) return;
    float* q = &p[(size_t)i * stride];
    *q = tanhf(*q);
}

// --------------------------------------------------------- layer 4 (32 -> 1)
__global__ void layer4_k(const float* __restrict__ x3, int stride,
                         const float* __restrict__ W4, const float* __restrict__ b4,
                         const float* __restrict__ self_norm,
                         float* __restrict__ h4,
                         float* __restrict__ out, int out_stride) {
    int i = blockIdx.x * blockDim.x + threadIdx.x;
    if (i >= NNODES) return;
    const float* r = x3 + (size_t)i * stride;
    float acc = 0.0f;
#pragma unroll
    for (int f = 0; f < 32; ++f) acc += r[f] * W4[f];
    h4[i] = acc;
    out[(size_t)i * out_stride] = acc * self_norm[i] + b4[0];
}

// -------------------------------------------------- per-graph node ranges
__global__ void init_bounds(int* __restrict__ gstart, int* __restrict__ gend) {
    int g = threadIdx.x;
    if (g < NGRAPH) { gstart[g] = NNODES; gend[g] = 0; }
}
__global__ void graph_bounds(const int* __restrict__ batch,
                             int* __restrict__ gstart, int* __restrict__ gend) {
    int i = blockIdx.x * blockDim.x + threadIdx.x;
    if (i >= NNODES) return;
    int b = batch[i];
    atomicMin(&gstart[b], i);
    atomicMax(&gend[b], i + 1);
}

// --------------------------------------------------------- per-graph top-K
// Strict (value, -index) threshold selection: matches top_k's descending
// order with ties broken toward the lower index.
__global__ __launch_bounds__(256)
void topk_kernel(const float* __restrict__ key, int stride,
                 const int* __restrict__ gstart, const int* __restrict__ gend,
                 int* __restrict__ topi, float* __restrict__ topv) {
    const int g = blockIdx.x, tid = threadIdx.x;
    __shared__ float sv[256];
    __shared__ int   si[256];
    __shared__ float shV;
    __shared__ int   shI;
    int s = gstart[g], e = gend[g];
    if (s > e) { s = 0; e = 0; }
    float curV = 0.0f; int curI = 0;
    for (int k = 0; k < KSEL; ++k) {
        float bv = -__builtin_inff();
        int   bi = 0x7fffffff;
        for (int i = s + tid; i < e; i += 256) {
            float v = key[(size_t)i * stride];
            bool elig = (k == 0) || (v < curV) || (v == curV && i > curI);
            if (elig && (v > bv || (v == bv && i < bi))) { bv = v; bi = i; }
        }
        sv[tid] = bv; si[tid] = bi;
        __syncthreads();
        for (int off = 128; off > 0; off >>= 1) {
            if (tid < off) {
                if (sv[tid + off] > sv[tid] ||
                    (sv[tid + off] == sv[tid] && si[tid + off] < si[tid])) {
                    sv[tid] = sv[tid + off]; si[tid] = si[tid + off];
                }
            }
            __syncthreads();
        }
        if (tid == 0) {
            if (si[0] == 0x7fffffff) {
                topv[g * KSEL + k] = -__builtin_inff();
                topi[g * KSEL + k] = 0;
                shV = -__builtin_inff(); shI = 0x7fffffff;
            } else {
                topv[g * KSEL + k] = sv[0];
                topi[g * KSEL + k] = si[0];
                shV = sv[0]; shI = si[0];
            }
        }
        __syncthreads();
        curV = shV; curI = shI;
    }
}

// ------------------------------------------------- conv/FC head, 1 block/graph
__global__ __launch_bounds__(128)
void head_kernel(const float* __restrict__ xc,
                 const int* __restrict__ topi, const float* __restrict__ topv,
                 const float* __restrict__ w5, const float* __restrict__ b5,
                 const float* __restrict__ w6, const float* __restrict__ b6,
                 const float* __restrict__ fw1, const float* __restrict__ fb1,
                 const float* __restrict__ fw2, const float* __restrict__ fb2,
                 float* __restrict__ outp) {
    const int g = blockIdx.x, tid = threadIdx.x;
    __shared__ float feats[KSEL * DCAT];  // 30 x 97
    __shared__ float c5[16 * KSEL];       // 16 x 30
    __shared__ float pool[16 * 15];
    __shared__ float c6[32 * 11];
    __shared__ float f1[128];
    __shared__ float logits[NCLS];

    for (int t = tid; t < KSEL * DCAT; t += 128) {
        int j = t / DCAT, f = t - j * DCAT;
        float tv = topv[g * KSEL + j];
        float val = 0.0f;
        if (tv > -1e30f) {                       // finite (keyf is a tanh value)
            int node = topi[g * KSEL + j];
            val = xc[(size_t)node * DCAT + f];
        }
        feats[t] = val;
    }
    __syncthreads();

    // conv5: kernel 97, stride 97 over flattened (K*97) == per-row dot
    for (int t = tid; t < 16 * KSEL; t += 128) {
        int o = t / KSEL, j = t - o * KSEL;
        float s = b5[o];
        for (int f = 0; f < DCAT; ++f) s += w5[o * DCAT + f] * feats[j * DCAT + f];
        c5[o * KSEL + j] = fmaxf(s, 0.0f);
    }
    __syncthreads();

    for (int t = tid; t < 16 * 15; t += 128) {
        int o = t / 15, q = t - o * 15;
        pool[t] = fmaxf(c5[o * KSEL + 2 * q], c5[o * KSEL + 2 * q + 1]);
    }
    __syncthreads();

    // conv6: 16 -> 32, kernel 5, length 15 -> 11
    for (int t = tid; t < 32 * 11; t += 128) {
        int o = t / 11, p = t - o * 11;
        float s = b6[o];
        for (int ic = 0; ic < 16; ++ic)
#pragma unroll
            for (int u = 0; u < 5; ++u)
                s += w6[(o * 16 + ic) * 5 + u] * pool[ic * 15 + p + u];
        c6[o * 11 + p] = fmaxf(s, 0.0f);
    }
    __syncthreads();

    if (tid < 128) {
        float s = fb1[tid];
        for (int i = 0; i < 352; ++i) s += c6[i] * fw1[i * 128 + tid];
        f1[tid] = fmaxf(s, 0.0f);
    }
    __syncthreads();

    if (tid < NCLS) {
        float s = fb2[tid];
        for (int i = 0; i < 128; ++i) s += f1[i] * fw2[i * NCLS + tid];
        logits[tid] = s;
    }
    __syncthreads();

    if (tid == 0) {
        float m = logits[0];
        for (int c = 1; c < NCLS; ++c) m = fmaxf(m, logits[c]);
        float lse = 0.0f;
        for (int c = 0; c < NCLS; ++c) lse += expf(logits[c] - m);
        lse = logf(lse);
        for (int c = 0; c < NCLS; ++c) outp[g * NCLS + c] = logits[c] - m - lse;
    }
}

// -------------------------------------------------------------------- launch
extern "C" void kernel_launch(void* const* d_in, const int* in_sizes, int n_in,
                              void* d_out, int out_size, void* d_ws, size_t ws_size,
                              hipStream_t stream) {
    const float* x       = (const float*)d_in[0];
    const int*   eidx    = (const int*)d_in[1];     // [2][E]
    const int*   batch   = (const int*)d_in[2];
    const float* W1      = (const float*)d_in[3];
    const float* b1      = (const float*)d_in[4];
    const float* W2      = (const float*)d_in[5];
    const float* b2      = (const float*)d_in[6];
    const float* W3      = (const float*)d_in[7];
    const float* b3      = (const float*)d_in[8];
    const float* W4      = (const float*)d_in[9];
    const float* b4      = (const float*)d_in[10];
    const float* conv5_w = (const float*)d_in[11];
    const float* conv5_b = (const float*)d_in[12];
    const float* conv6_w = (const float*)d_in[13];
    const float* conv6_b = (const float*)d_in[14];
    const float* fc1_w   = (const float*)d_in[15];
    const float* fc1_b   = (const float*)d_in[16];
    const float* fc2_w   = (const float*)d_in[17];
    const float* fc2_b   = (const float*)d_in[18];
    float* out = (float*)d_out;

    const int* src = eidx;
    const int* dst = eidx + NEDGES;

    // ---- workspace carve-out (~119 MB)
    char* p = (char*)d_ws;
    float* deg   = (float*)p; p += (size_t)NNODES * 4;            // -> self_norm
    float* dinv  = (float*)p; p += (size_t)NNODES * 4;
    float* enorm = (float*)p; p += (size_t)NEDGES * 4;
    float* h     = (float*)p; p += (size_t)NNODES * 32 * 4;
    float* h4    = (float*)p; p += (size_t)NNODES * 4;
    float* xc    = (float*)p; p += (size_t)NNODES * DCAT * 4;
    int*   gstart = (int*)p;  p += (size_t)NGRAPH * 4;
    int*   gend   = (int*)p;  p += (size_t)NGRAPH * 4;
    int*   topi   = (int*)p;  p += (size_t)NGRAPH * KSEL * 4;
    float* topv  = (float*)p; p += (size_t)NGRAPH * KSEL * 4;
    (void)ws_size; (void)n_in; (void)in_sizes; (void)out_size;

    auto cdiv = [](long long a, long long b) { return (unsigned)((a + b - 1) / b); };

    // ---- graph normalization
    fill_f32<<<cdiv(NNODES, 256), 256, 0, stream>>>(deg, 1.0f, NNODES);
    deg_accum<<<cdiv(NEDGES, 256), 256, 0, stream>>>(src, dst, deg);
    norms_k<<<cdiv(NNODES, 256), 256, 0, stream>>>(deg, dinv);   // deg -> self_norm
    enorm_k<<<cdiv(NEDGES, 256), 256, 0, stream>>>(src, dst, dinv, enorm);

    const unsigned gemmBlocks = NNODES / 64;                     // 3125 (2 waves/blk)
    const unsigned scatBlocks = cdiv((long long)NEDGES * 32, 256);
    const unsigned tanhBlocks = cdiv((long long)NNODES * 32, 256);

    // ---- GCN layer 1 (128 -> 32) into xc[:, 0:32]
    gemm_wmma_f32<<<gemmBlocks, 64, 0, stream>>>(x, FIN, W1, FIN, b1, deg, h, xc + 0, DCAT);
    scatter32<<<scatBlocks, 256, 0, stream>>>(src, dst, enorm, h, xc + 0, DCAT);
    tanh32<<<tanhBlocks, 256, 0, stream>>>(xc + 0, DCAT);

    // ---- GCN layer 2 (32 -> 32) into xc[:, 32:64]
    gemm_wmma_f32<<<gemmBlocks, 64, 0, stream>>>(xc + 0, DCAT, W2, 32, b2, deg, h, xc + 32, DCAT);
    scatter32<<<scatBlocks, 256, 0, stream>>>(src, dst, enorm, h, xc + 32, DCAT);
    tanh32<<<tanhBlocks, 256, 0, stream>>>(xc + 32, DCAT);

    // ---- GCN layer 3 (32 -> 32) into xc[:, 64:96]
    gemm_wmma_f32<<<gemmBlocks, 64, 0, stream>>>(xc + 32, DCAT, W3, 32, b3, deg, h, xc + 64, DCAT);
    scatter32<<<scatBlocks, 256, 0, stream>>>(src, dst, enorm, h, xc + 64, DCAT);
    tanh32<<<tanhBlocks, 256, 0, stream>>>(xc + 64, DCAT);

    // ---- GCN layer 4 (32 -> 1) into xc[:, 96]
    layer4_k<<<cdiv(NNODES, 256), 256, 0, stream>>>(xc + 64, DCAT, W4, b4, deg, h4, xc + 96, DCAT);
    scatter1<<<cdiv(NEDGES, 256), 256, 0, stream>>>(src, dst, enorm, h4, xc + 96, DCAT);
    tanh1<<<cdiv(NNODES, 256), 256, 0, stream>>>(xc + 96, DCAT);

    // ---- per-graph top-K on keyf = xc[:, 96]
    init_bounds<<<1, 128, 0, stream>>>(gstart, gend);
    graph_bounds<<<cdiv(NNODES, 256), 256, 0, stream>>>(batch, gstart, gend);
    topk_kernel<<<NGRAPH, 256, 0, stream>>>(xc + 96, DCAT, gstart, gend, topi, topv);

    // ---- conv/FC head + log_softmax
    head_kernel<<<NGRAPH, 128, 0, stream>>>(xc, topi, topv,
                                            conv5_w, conv5_b, conv6_w, conv6_b,
                                            fc1_w, fc1_b, fc2_w, fc2_b, out);
}